// DenseGCM_25872882991587
// MI455X (gfx1250) — compile-verified
//
#include <hip/hip_runtime.h>
#include <math.h>

#define BB 64
#define NN 1024
#define FF 256

typedef __attribute__((ext_vector_type(2))) float v2f;
typedef __attribute__((ext_vector_type(4))) float v4f;
typedef __attribute__((ext_vector_type(8))) float v8f;

// ---------------------------------------------------------------------------
// Kernel 1: nodes wrap (FIFO eviction) + scatter of x -> nodes_out
// Regular (RT) cache policy: nodes_out (67MB) is re-read by k_gemv and fits
// in the 192MB L2, so we want it resident.
// ---------------------------------------------------------------------------
__global__ void k_nodes(const float* __restrict__ nodes, const float* __restrict__ x,
                        const int* __restrict__ num, float* __restrict__ nodes_out) {
    long idx = (long)blockIdx.x * blockDim.x + threadIdx.x;        // float4 index
    const long total = (long)BB * NN * (FF / 4);
    if (idx >= total) return;
    int b  = (int)(idx / (NN * (FF / 4)));
    int r  = (int)(idx % (NN * (FF / 4)));
    int i  = r / (FF / 4);
    int f4 = (r % (FF / 4)) * 4;
    int nm = num[b];
    bool full = (nm == NN);
    float4 v;
    if (full) {
        if (i == NN - 1) v = *(const float4*)(x + (long)b * FF + f4);
        else             v = *(const float4*)(nodes + ((long)b * NN + i + 1) * FF + f4);
    } else {
        if (i == nm)     v = *(const float4*)(x + (long)b * FF + f4);
        else             v = *(const float4*)(nodes + ((long)b * NN + i) * FF + f4);
    }
    *(float4*)(nodes_out + ((long)b * NN + i) * FF + f4) = v;
}

// ---------------------------------------------------------------------------
// Kernel 2: adj wrap -> adj_out
// 268MB read + 268MB write of stream-once data. Non-temporal (TH=NT) loads
// and stores so this stream does not evict the reused nodes_out working set
// from L2 (CDNA5 cache temporal hints, ISA ch.7.3).
// ---------------------------------------------------------------------------
__global__ void k_adj(const float* __restrict__ adj, const int* __restrict__ num,
                      float* __restrict__ adj_out) {
    long idx = (long)blockIdx.x * blockDim.x + threadIdx.x;        // float4 index
    const long total = (long)BB * NN * (NN / 4);
    if (idx >= total) return;
    int b  = (int)(idx / (NN * (NN / 4)));
    int r  = (int)(idx % (NN * (NN / 4)));
    int i  = r / (NN / 4);
    int j4 = (r % (NN / 4)) * 4;
    bool full = (num[b] == NN);
    const float* src = adj + (long)b * NN * NN;
    v4f v;
    if (!full) {
        v = __builtin_nontemporal_load((const v4f*)(src + (long)i * NN + j4));
    } else if (i == NN - 1) {
        v = (v4f){0.f, 0.f, 0.f, 0.f};
    } else {
        float e[4];
#pragma unroll
        for (int t = 0; t < 4; ++t) {
            int j = j4 + t;
            e[t] = (j < NN - 1) ? src[(long)(i + 1) * NN + j + 1] : 0.f;
        }
        v = (v4f){e[0], e[1], e[2], e[3]};
    }
    __builtin_nontemporal_store(v, (v4f*)(adj_out + (long)b * NN * NN + (long)i * NN + j4));
}

// ---------------------------------------------------------------------------
// Kernel 3: num_nodes output (post-wrap num + 1), stored as float
// ---------------------------------------------------------------------------
__global__ void k_num(const int* __restrict__ num, float* __restrict__ num_out) {
    int b = threadIdx.x;
    if (b < BB) {
        int nm = num[b];
        int ne = (nm == NN) ? NN - 1 : nm;
        num_out[b] = (float)(ne + 1);
    }
}

// ---------------------------------------------------------------------------
// Kernel 4: batched GEMV  t[b,:] = adj_new[b, r, :] @ nodes_new[b]   (r = slot)
// One block per batch; adj row staged in LDS; 256 threads = 256 features.
// Memory-bound (reads all of nodes once, hopefully L2-resident) -> VALU FMAs.
// ---------------------------------------------------------------------------
__global__ void __launch_bounds__(256) k_gemv(const float* __restrict__ nodes_new,
                                              const float* __restrict__ adj_new,
                                              const int* __restrict__ num,
                                              float* __restrict__ T) {
    int b  = blockIdx.x;
    int nm = num[b];
    int r  = (nm == NN) ? NN - 1 : nm;
    __shared__ float arow[NN];
    for (int k = threadIdx.x; k < NN; k += blockDim.x)
        arow[k] = adj_new[((long)b * NN + r) * NN + k];
    __syncthreads();
    int f = threadIdx.x;
    const float* np = nodes_new + (long)b * NN * FF + f;
    float acc = 0.f;
#pragma unroll 8
    for (int k = 0; k < NN; ++k)
        acc = fmaf(arow[k], np[(long)k * FF], acc);
    T[b * FF + f] = acc;
}

// ---------------------------------------------------------------------------
// Kernel 5: mx = tanh(T @ W)  with  T[64x256], W[256x256]  via
// V_WMMA_F32_16X16X4_F32. Grid = 4 blocks (16 batch-rows each), 8 waves/block,
// each wave owns a 16x32 tile (2 accumulators). K loop: 256/4 = 64 iterations.
// Operand layout per CDNA5 ISA 7.12.2 (fp32 A 16x4: lanes 0-15 K={k,k+1},
// lanes 16-31 K={k+2,k+3}; B 4x16 mirrored; C/D: VGPR v <-> row v / v+8).
// ---------------------------------------------------------------------------
__global__ void __launch_bounds__(256) k_gemm(const float* __restrict__ T,
                                              const float* __restrict__ W,
                                              float* __restrict__ mx) {
    int m0   = blockIdx.x * 16;
    int wave = threadIdx.x >> 5;
    int lane = threadIdx.x & 31;
    int hi   = lane >> 4;          // 0: lanes 0-15, 1: lanes 16-31
    int lm   = lane & 15;
    int c0   = wave * 32;
    v8f acc0 = {};
    v8f acc1 = {};
    for (int k = 0; k < FF; k += 4) {
        int ka = k + 2 * hi;       // even -> 8B-aligned float2 load
        v2f a = *(const v2f*)(T + (m0 + lm) * FF + ka);
        v2f b0, b1;
        b0[0] = W[ka * FF + c0 + lm];
        b0[1] = W[(ka + 1) * FF + c0 + lm];
        b1[0] = W[ka * FF + c0 + 16 + lm];
        b1[1] = W[(ka + 1) * FF + c0 + 16 + lm];
        acc0 = __builtin_amdgcn_wmma_f32_16x16x4_f32(false, a, false, b0,
                                                     (short)0, acc0, false, false);
        acc1 = __builtin_amdgcn_wmma_f32_16x16x4_f32(false, a, false, b1,
                                                     (short)0, acc1, false, false);
    }
#pragma unroll
    for (int v = 0; v < 8; ++v) {
        int row = m0 + v + 8 * hi;
        mx[row * FF + c0 + lm]      = tanhf(acc0[v]);
        mx[row * FF + c0 + 16 + lm] = tanhf(acc1[v]);
    }
}

// ---------------------------------------------------------------------------
extern "C" void kernel_launch(void* const* d_in, const int* in_sizes, int n_in,
                              void* d_out, int out_size, void* d_ws, size_t ws_size,
                              hipStream_t stream) {
    const float* x     = (const float*)d_in[0];
    const float* nodes = (const float*)d_in[1];
    const float* adj   = (const float*)d_in[2];
    const int*   num   = (const int*)d_in[3];
    const float* W     = (const float*)d_in[4];

    float* out       = (float*)d_out;
    float* mx_out    = out;                                   // [64 x 256]
    float* nodes_out = out + (long)BB * FF;                   // [64 x 1024 x 256]
    float* adj_out   = nodes_out + (long)BB * NN * FF;        // [64 x 1024 x 1024]
    float* num_out   = adj_out + (long)BB * NN * NN;          // [64]
    float* T         = (float*)d_ws;                          // [64 x 256] scratch

    {   // nodes wrap + scatter (RT policy: re-read by k_gemv, keep in L2)
        long n4 = (long)BB * NN * (FF / 4);
        k_nodes<<<(int)((n4 + 255) / 256), 256, 0, stream>>>(nodes, x, num, nodes_out);
    }
    {   // adj wrap (NT policy: stream-once, don't thrash L2)
        long n4 = (long)BB * NN * (NN / 4);
        k_adj<<<(int)((n4 + 255) / 256), 256, 0, stream>>>(adj, num, adj_out);
    }
    k_num<<<1, 64, 0, stream>>>(num, num_out);

    // batched GEMV over updated tensors (reads back from d_out regions)
    k_gemv<<<BB, 256, 0, stream>>>(nodes_out, adj_out, num, T);

    // fp32 WMMA GEMM + tanh epilogue -> mx
    k_gemm<<<BB / 16, 256, 0, stream>>>(T, W, mx_out);
}